// GE2ELoss_1580547974046
// MI455X (gfx1250) — compile-verified
//
#include <hip/hip_runtime.h>
#include <hip/hip_bf16.h>

#define EPS_COS 1e-08f
#define EPS_SIM 1e-06f
#define N_SPK   1024
#define M_UTT   10
#define D_DIM   256
#define R_ROWS  (N_SPK * M_UTT)   // 10240

typedef float v2f __attribute__((ext_vector_type(2)));
typedef float v4f __attribute__((ext_vector_type(4)));
typedef float v8f __attribute__((ext_vector_type(8)));

__global__ void ge2e_zero(float* out) { out[0] = 0.0f; }

// ---------------------------------------------------------------------------
// Kernel 1: per-speaker prep.
// grid = N_SPK blocks, 256 threads (= D).  Thread tid handles dimension d=tid.
// Produces: centT[d][k] (D x N, so WMMA B-loads are row-contiguous in k),
//           cent_norm[k], emb_norm[row], pos_vals[row] = w*(cos_same+eps)+b.
// ---------------------------------------------------------------------------
__global__ __launch_bounds__(256)
void ge2e_prep(const float* __restrict__ emb,
               const float* __restrict__ wp, const float* __restrict__ bp,
               float* __restrict__ centT, float* __restrict__ cent_norm,
               float* __restrict__ emb_norm, float* __restrict__ pos_vals)
{
    __shared__ float lds_part[8 * 31];
    __shared__ float lds_tot[31];

    const int n   = blockIdx.x;
    const int tid = threadIdx.x;          // = d
    const float* row0 = emb + (size_t)n * M_UTT * D_DIM;

    float e[M_UTT];
    float s = 0.0f;
#pragma unroll
    for (int m = 0; m < M_UTT; ++m) { e[m] = row0[m * D_DIM + tid]; s += e[m]; }

    const float c = s * (1.0f / (float)M_UTT);
    centT[(size_t)tid * N_SPK + n] = c;

    // 31 reduction values: [0..9] dot(e,u), [10..19] |e|^2, [20..29] |u|^2, [30] |c|^2
    float part[31];
    part[30] = c * c;
#pragma unroll
    for (int m = 0; m < M_UTT; ++m) {
        const float u = (s - e[m]) * (1.0f / (float)(M_UTT - 1));
        part[m]      = e[m] * u;
        part[10 + m] = e[m] * e[m];
        part[20 + m] = u * u;
    }

    // wave32 reduction of all 31 values
#pragma unroll
    for (int v = 0; v < 31; ++v) {
        float r = part[v];
#pragma unroll
        for (int off = 16; off > 0; off >>= 1) r += __shfl_xor(r, off);
        part[v] = r;
    }
    const int lane = tid & 31, wave = tid >> 5;
    if (lane == 0) {
#pragma unroll
        for (int v = 0; v < 31; ++v) lds_part[wave * 31 + v] = part[v];
    }
    __syncthreads();
    if (tid < 31) {
        float t = 0.0f;
#pragma unroll
        for (int w8 = 0; w8 < 8; ++w8) t += lds_part[w8 * 31 + tid];
        lds_tot[tid] = t;
    }
    __syncthreads();

    if (tid < M_UTT) {
        const float W  = wp[0], Bb = bp[0];
        const int row  = n * M_UTT + tid;
        const float dt = lds_tot[tid];
        const float en = sqrtf(lds_tot[10 + tid]);
        const float un = sqrtf(lds_tot[20 + tid]);
        const float cs = dt / fmaxf(en * un, EPS_COS);
        emb_norm[row] = en;
        pos_vals[row] = W * (cs + EPS_SIM) + Bb;
    }
    if (tid == 0) cent_norm[n] = sqrtf(lds_tot[30]);
}

// fast reciprocal: v_rcp_f32 + one Newton step (~0.5 ulp)
__device__ __forceinline__ float fast_rcp(float x) {
    float inv = __builtin_amdgcn_rcpf(x);
    return __builtin_fmaf(__builtin_fmaf(-x, inv, 1.0f), inv, inv);
}

// ---------------------------------------------------------------------------
// Kernel 2: fused WMMA GEMM (10240 x 1024, K=256) + row-wise log-sum-exp.
// grid = R_ROWS/16 = 640 blocks, 256 threads = 8 waves.
// Block: 16-row A panel in LDS (stride 260 -> 32 lanes x b64 hit all 64 banks;
// 260*4 = 1040 is 16B-aligned so staging can use b128);
// wave w covers columns [w*128, w*128+128) as 8 col-tiles of 16,
// each col-tile = 64 x V_WMMA_F32_16X16X4_F32 over K=256.
// Epilogue: sim = fma(dot*inv_cn, w*inv_en[i], w*eps+b), diag via cndmask,
// __expf accumulate; hierarchical reduction -> log -> atomicAdd.
// ---------------------------------------------------------------------------
__global__ __launch_bounds__(256)
void ge2e_gemm_lse(const float* __restrict__ emb,
                   const float* __restrict__ centT,
                   const float* __restrict__ cent_norm,
                   const float* __restrict__ emb_norm,
                   const float* __restrict__ pos_vals,
                   const float* __restrict__ wp, const float* __restrict__ bp,
                   float* __restrict__ out)
{
    __shared__ float lds_A[16][260];
    __shared__ float lds_rowsum[16];
    __shared__ float lds_embn[16];
    __shared__ float lds_pos[16];

    const int tid     = threadIdx.x;
    const int rowBase = blockIdx.x * 16;

    // stage A panel: 16 rows x 256 dims, vectorized 128-bit
#pragma unroll
    for (int i = 0; i < 4; ++i) {
        const int f  = tid + i * 256;      // float4 index, 1024 total
        const int r  = f >> 6;             // 64 float4 per row
        const int d4 = f & 63;
        const v4f vsrc = *(const v4f*)(emb + (size_t)(rowBase + r) * D_DIM + d4 * 4);
        *(v4f*)(&lds_A[r][d4 * 4]) = vsrc;
    }
    if (tid < 16) {
        lds_rowsum[tid] = 0.0f;
        lds_embn[tid]   = emb_norm[rowBase + tid];
        lds_pos[tid]    = pos_vals[rowBase + tid];
    }
    __syncthreads();

    const float W = wp[0], Bb = bp[0];
    const float simc = __builtin_fmaf(W, EPS_SIM, Bb);   // w*eps + b
    const int wave = tid >> 5;
    const int lane = tid & 31;
    const int half = lane >> 4;          // 0: K%4 in {0,1} / rows 0-7 ; 1: {2,3} / rows 8-15
    const int l16  = lane & 15;

    // hoist per-row constants: wi = w / emb_norm, pos, diagonal column, accum
    float wi[8], posv[8], rs[8];
    int nrowv[8];
#pragma unroll
    for (int i = 0; i < 8; ++i) {
        const int rl = i + half * 8;
        wi[i]    = W * fast_rcp(fmaxf(lds_embn[rl], 1e-30f));
        posv[i]  = lds_pos[rl];
        nrowv[i] = (rowBase + rl) / M_UTT;   // diagonal column index for this row
        rs[i]    = 0.0f;
    }

    for (int ct = 0; ct < 8; ++ct) {
        const int k = wave * 128 + ct * 16 + l16;   // this lane's output column
        v8f acc = {};
        const float* aptr = &lds_A[l16][half * 2];
        const float* bptr = centT + (size_t)(half * 2) * N_SPK + k;
#pragma unroll 8
        for (int d = 0; d < D_DIM; d += 4) {
            v2f a, b;
            a.x = aptr[d];
            a.y = aptr[d + 1];
            b.x = bptr[(size_t)d * N_SPK];
            b.y = bptr[(size_t)(d + 1) * N_SPK];
            acc = __builtin_amdgcn_wmma_f32_16x16x4_f32(
                /*neg_a=*/false, a, /*neg_b=*/false, b,
                /*c_mod=*/(short)0, acc, /*reuse_a=*/false, /*reuse_b=*/false);
        }
        const float inv_cn = fast_rcp(fmaxf(cent_norm[k], 1e-30f));
#pragma unroll
        for (int i = 0; i < 8; ++i) {
            float sim = __builtin_fmaf(acc[i] * inv_cn, wi[i], simc);
            sim = (k == nrowv[i]) ? posv[i] : sim;   // diagonal -> cos_same (cndmask)
            rs[i] += __expf(sim);
        }
    }

    // reduce Σexp across the 16 lanes that share each row
#pragma unroll
    for (int i = 0; i < 8; ++i) {
        float v = rs[i];
        v += __shfl_xor(v, 1); v += __shfl_xor(v, 2);
        v += __shfl_xor(v, 4); v += __shfl_xor(v, 8);
        if (l16 == 0) atomicAdd(&lds_rowsum[i + half * 8], v);
    }
    __syncthreads();

    if (tid < 16) {
        float part = logf(lds_rowsum[tid] + EPS_SIM) - lds_pos[tid];
        part += __shfl_xor(part, 1); part += __shfl_xor(part, 2);
        part += __shfl_xor(part, 4); part += __shfl_xor(part, 8);
        if (tid == 0) atomicAdd(out, part);
    }
}

extern "C" void kernel_launch(void* const* d_in, const int* in_sizes, int n_in,
                              void* d_out, int out_size, void* d_ws, size_t ws_size,
                              hipStream_t stream) {
    const float* emb = (const float*)d_in[0];
    const float* w   = (const float*)d_in[1];
    const float* b   = (const float*)d_in[2];
    float* out = (float*)d_out;

    float* ws        = (float*)d_ws;
    float* centT     = ws;                                   // D*N   = 262144 floats
    float* cent_norm = ws + (size_t)D_DIM * N_SPK;           // N     =   1024
    float* emb_norm  = cent_norm + N_SPK;                    // R     =  10240
    float* pos_vals  = emb_norm + R_ROWS;                    // R     =  10240

    ge2e_zero<<<1, 1, 0, stream>>>(out);
    ge2e_prep<<<N_SPK, 256, 0, stream>>>(emb, w, b, centT, cent_norm, emb_norm, pos_vals);
    ge2e_gemm_lse<<<R_ROWS / 16, 256, 0, stream>>>(emb, centT, cent_norm, emb_norm,
                                                   pos_vals, w, b, out);
}